// PAM_Module_52029233824282
// MI455X (gfx1250) — compile-verified
//
#include <hip/hip_runtime.h>
#include <hip/hip_bf16.h>
#include <math.h>

// PAM module for MI455X (gfx1250, wave32).
// Shapes: B=32, C=1024, H=W=32, N=H*W=1024, C8=128.
// Memory-bound op (~94 GFLOP vs ~0.9 GB traffic) -> keep fp32, use
// V_WMMA_F32_16X16X4_F32 for all GEMMs.
//
// Workspace layout (floats):
//   Qbuf: [B][C8][N]   = 4,194,304 floats (16 MB)
//   Kbuf: [B][C8][N]   = 4,194,304 floats (16 MB)
//   Ebuf: [B][N][N]    = 33,554,432 floats (128 MB)  (energy -> attention in place)

typedef __attribute__((ext_vector_type(2))) float v2f;
typedef __attribute__((ext_vector_type(8))) float v8f;

#define WMMA_F32(a, b, c) \
  __builtin_amdgcn_wmma_f32_16x16x4_f32(false, (a), false, (b), (short)0, (c), false, false)

// ---------------------------------------------------------------------------
// Kernel 1: Q = Wq @ X + bq ; K = Wk @ X + bk      (blockIdx.z selects Q/K)
// Wq/Wk: [128 x 1024] row-major, X: [1024 x 1024] row-major per batch.
// Block = 128 threads = 4 waves; each wave owns a 16x64 tile; block = 64x64.
// ---------------------------------------------------------------------------
__global__ __launch_bounds__(128)
void proj_qk_kernel(const float* __restrict__ x,
                    const float* __restrict__ Wq, const float* __restrict__ bq,
                    const float* __restrict__ Wk, const float* __restrict__ bk,
                    float* __restrict__ Qbuf, float* __restrict__ Kbuf)
{
    const int N = 1024, C = 1024, C8 = 128;
    const int b = blockIdx.y;
    const float* W;
    const float* bias;
    float* Out;
    if (blockIdx.z == 0) { W = Wq; bias = bq; Out = Qbuf; }
    else                 { W = Wk; bias = bk; Out = Kbuf; }
    const float* Xb = x + (size_t)b * C * N;     // [C x N]
    Out += (size_t)b * C8 * N;

    const int tile  = blockIdx.x;    // 0..31 : 2 M-blocks x 16 N-blocks
    const int mblk  = tile >> 4;
    const int nblk  = tile & 15;
    const int wave  = threadIdx.x >> 5;
    const int lane  = threadIdx.x & 31;
    const int lhalf = lane >> 4;     // 0 or 1
    const int lmod  = lane & 15;

    const int m0 = mblk * 64 + wave * 16;
    const int n0 = nblk * 64;

    v8f acc0 = {}, acc1 = {}, acc2 = {}, acc3 = {};
    const float* arow = W + (size_t)(m0 + lmod) * C;   // A row for this lane
    for (int k0 = 0; k0 < C; k0 += 4) {
        const int ka = k0 + 2 * lhalf;
        v2f a = *(const v2f*)(arow + ka);              // A[M=lmod][K=ka, ka+1]
        const float* bp = Xb + (size_t)ka * N + n0 + lmod;
        v2f bb;
        bb.x = bp[0];      bb.y = bp[N];      acc0 = WMMA_F32(a, bb, acc0);
        bb.x = bp[16];     bb.y = bp[N + 16]; acc1 = WMMA_F32(a, bb, acc1);
        bb.x = bp[32];     bb.y = bp[N + 32]; acc2 = WMMA_F32(a, bb, acc2);
        bb.x = bp[48];     bb.y = bp[N + 48]; acc3 = WMMA_F32(a, bb, acc3);
    }
#pragma unroll
    for (int r = 0; r < 8; ++r) {
        const int row = m0 + r + 8 * lhalf;
        const float bs = bias[row];
        float* o = Out + (size_t)row * N + n0 + lmod;
        o[0]  = acc0[r] + bs;
        o[16] = acc1[r] + bs;
        o[32] = acc2[r] + bs;
        o[48] = acc3[r] + bs;
    }
}

// ---------------------------------------------------------------------------
// Kernel 2: E[n,m] = sum_o Q[o,n] * K[o,m]   (K-dim = 128)
// Q, K stored [C8 x N]; A is read "transposed" (column-wise), which matches
// the same lane pattern as B.
// ---------------------------------------------------------------------------
__global__ __launch_bounds__(128)
void energy_kernel(const float* __restrict__ Qbuf,
                   const float* __restrict__ Kbuf,
                   float* __restrict__ Ebuf)
{
    const int N = 1024, C8 = 128;
    const int b = blockIdx.y;
    const float* Q  = Qbuf + (size_t)b * C8 * N;
    const float* Kp = Kbuf + (size_t)b * C8 * N;
    float* E = Ebuf + (size_t)b * N * N;

    const int tile  = blockIdx.x;    // 0..255 : 16 x 16
    const int mblk  = tile >> 4;
    const int nblk  = tile & 15;
    const int wave  = threadIdx.x >> 5;
    const int lane  = threadIdx.x & 31;
    const int lhalf = lane >> 4;
    const int lmod  = lane & 15;

    const int m0 = mblk * 64 + wave * 16;   // energy row (n)
    const int n0 = nblk * 64;               // energy col (m)

    v8f acc0 = {}, acc1 = {}, acc2 = {}, acc3 = {};
    for (int k0 = 0; k0 < C8; k0 += 4) {
        const int ka = k0 + 2 * lhalf;
        const float* qp = Q + (size_t)ka * N + m0 + lmod;
        v2f a;  a.x = qp[0];  a.y = qp[N];          // A[M][K] = Q[K][M]
        const float* kp = Kp + (size_t)ka * N + n0 + lmod;
        v2f bb;
        bb.x = kp[0];   bb.y = kp[N];      acc0 = WMMA_F32(a, bb, acc0);
        bb.x = kp[16];  bb.y = kp[N + 16]; acc1 = WMMA_F32(a, bb, acc1);
        bb.x = kp[32];  bb.y = kp[N + 32]; acc2 = WMMA_F32(a, bb, acc2);
        bb.x = kp[48];  bb.y = kp[N + 48]; acc3 = WMMA_F32(a, bb, acc3);
    }
#pragma unroll
    for (int r = 0; r < 8; ++r) {
        const int row = m0 + r + 8 * lhalf;
        float* o = E + (size_t)row * N + n0 + lmod;
        o[0]  = acc0[r];
        o[16] = acc1[r];
        o[32] = acc2[r];
        o[48] = acc3[r];
    }
}

// ---------------------------------------------------------------------------
// Kernel 3: in-place row softmax over last axis (row length 1024).
// One 256-thread block per (b, n) row.
// ---------------------------------------------------------------------------
__global__ __launch_bounds__(256)
void softmax_rows_kernel(float* __restrict__ E)
{
    const int N = 1024;
    float* row = E + (size_t)blockIdx.x * N;
    __shared__ float red[256];
    const int t = threadIdx.x;

    float m = -INFINITY;
    for (int i = t; i < N; i += 256) m = fmaxf(m, row[i]);
    red[t] = m;
    __syncthreads();
    for (int s = 128; s > 0; s >>= 1) {
        if (t < s) red[t] = fmaxf(red[t], red[t + s]);
        __syncthreads();
    }
    m = red[0];
    __syncthreads();

    float sum = 0.0f;
    for (int i = t; i < N; i += 256) {
        const float e = expf(row[i] - m);
        row[i] = e;
        sum += e;
    }
    red[t] = sum;
    __syncthreads();
    for (int s = 128; s > 0; s >>= 1) {
        if (t < s) red[t] += red[t + s];
        __syncthreads();
    }
    const float inv = 1.0f / red[0];
    for (int i = t; i < N; i += 256) row[i] *= inv;
}

// ---------------------------------------------------------------------------
// Kernel 4: v = Wv @ X + bv (WMMA), fused epilogue:
//   out[b,c,n] = gamma * ((v[b,c,n]) * att[b,n,c]) + x[b,c,n]
// ---------------------------------------------------------------------------
__global__ __launch_bounds__(128)
void v_attn_out_kernel(const float* __restrict__ x,
                       const float* __restrict__ Wv, const float* __restrict__ bv,
                       const float* __restrict__ Att,
                       const float* __restrict__ gamma,
                       float* __restrict__ out)
{
    const int N = 1024, C = 1024;
    const int b = blockIdx.y;
    const float* Xb  = x + (size_t)b * C * N;
    const float* att = Att + (size_t)b * N * N;
    const float g = gamma[0];

    const int tile  = blockIdx.x;    // 0..255 : 16 x 16
    const int mblk  = tile >> 4;
    const int nblk  = tile & 15;
    const int wave  = threadIdx.x >> 5;
    const int lane  = threadIdx.x & 31;
    const int lhalf = lane >> 4;
    const int lmod  = lane & 15;

    const int m0 = mblk * 64 + wave * 16;   // channel c
    const int n0 = nblk * 64;               // spatial n

    v8f acc0 = {}, acc1 = {}, acc2 = {}, acc3 = {};
    const float* arow = Wv + (size_t)(m0 + lmod) * C;
    for (int k0 = 0; k0 < C; k0 += 4) {
        const int ka = k0 + 2 * lhalf;
        v2f a = *(const v2f*)(arow + ka);
        const float* bp = Xb + (size_t)ka * N + n0 + lmod;
        v2f bb;
        bb.x = bp[0];   bb.y = bp[N];      acc0 = WMMA_F32(a, bb, acc0);
        bb.x = bp[16];  bb.y = bp[N + 16]; acc1 = WMMA_F32(a, bb, acc1);
        bb.x = bp[32];  bb.y = bp[N + 32]; acc2 = WMMA_F32(a, bb, acc2);
        bb.x = bp[48];  bb.y = bp[N + 48]; acc3 = WMMA_F32(a, bb, acc3);
    }

    const int n = n0 + lmod;
#pragma unroll
    for (int r = 0; r < 8; ++r) {
        const int c = m0 + r + 8 * lhalf;
        const float bias = bv[c];
        const float* xrow = Xb + (size_t)c * N + n0 + lmod;
        float* orow = out + (size_t)b * C * N + (size_t)c * N + n0 + lmod;
        const float a0 = att[(size_t)(n)      * N + c];
        const float a1 = att[(size_t)(n + 16) * N + c];
        const float a2 = att[(size_t)(n + 32) * N + c];
        const float a3 = att[(size_t)(n + 48) * N + c];
        orow[0]  = g * ((acc0[r] + bias) * a0) + xrow[0];
        orow[16] = g * ((acc1[r] + bias) * a1) + xrow[16];
        orow[32] = g * ((acc2[r] + bias) * a2) + xrow[32];
        orow[48] = g * ((acc3[r] + bias) * a3) + xrow[48];
    }
}

// ---------------------------------------------------------------------------
extern "C" void kernel_launch(void* const* d_in, const int* in_sizes, int n_in,
                              void* d_out, int out_size, void* d_ws, size_t ws_size,
                              hipStream_t stream)
{
    (void)in_sizes; (void)n_in; (void)out_size; (void)ws_size;
    const float* x     = (const float*)d_in[0];   // [32,1024,32,32]
    const float* Wq    = (const float*)d_in[1];   // [128,1024]
    const float* bq    = (const float*)d_in[2];   // [128]
    const float* Wk    = (const float*)d_in[3];   // [128,1024]
    const float* bk    = (const float*)d_in[4];   // [128]
    const float* Wv    = (const float*)d_in[5];   // [1024,1024]
    const float* bv    = (const float*)d_in[6];   // [1024]
    const float* gamma = (const float*)d_in[7];   // [1]
    float* out = (float*)d_out;

    const int B = 32, C8 = 128, N = 1024;
    float* ws   = (float*)d_ws;
    float* Qbuf = ws;                                  // B*C8*N
    float* Kbuf = Qbuf + (size_t)B * C8 * N;           // B*C8*N
    float* Ebuf = Kbuf + (size_t)B * C8 * N;           // B*N*N

    dim3 blk(128);
    proj_qk_kernel<<<dim3(32, B, 2), blk, 0, stream>>>(x, Wq, bq, Wk, bk, Qbuf, Kbuf);
    energy_kernel<<<dim3(256, B), blk, 0, stream>>>(Qbuf, Kbuf, Ebuf);
    softmax_rows_kernel<<<dim3(B * N), dim3(256), 0, stream>>>(Ebuf);
    v_attn_out_kernel<<<dim3(256, B), blk, 0, stream>>>(x, Wv, bv, Ebuf, gamma, out);
}